// DDTreeSampler_59115929862476
// MI455X (gfx1250) — compile-verified
//
#include <hip/hip_runtime.h>
#include <hip/hip_bf16.h>
#include <stdint.h>

#define AS1 __attribute__((address_space(1)))
#define AS3 __attribute__((address_space(3)))

typedef int v4i_ __attribute__((vector_size(16)));

#if defined(__gfx1250__) && __has_builtin(__builtin_amdgcn_global_load_async_to_lds_b128)
#define USE_ASYNC_LDS 1
// global -> LDS async B128 copy (one 16B element per active lane)
#define GASYNC_B128(gp, lp)                                                 \
    __builtin_amdgcn_global_load_async_to_lds_b128(                         \
        (AS1 v4i_*)(void*)(gp), (AS3 v4i_*)(void*)(lp), 0, 0)
#else
#define USE_ASYNC_LDS 0
#endif

// ---------------------------------------------------------------------------
// Kernel 1: per-row argmax over V columns (memory-bound hot path).
// One 256-thread block per logits row; gathers via target_logits_indices.
// First-occurrence tie-break to match jnp.argmax.
// ---------------------------------------------------------------------------
__global__ void __launch_bounds__(256)
ddts_argmax_rows(const float* __restrict__ logits,
                 const int*   __restrict__ tgt_idx,
                 int*         __restrict__ posterior,
                 int V)
{
    const int row = blockIdx.x;
    const int tid = threadIdx.x;

    const long long src  = (long long)tgt_idx[row];
    const float*    base = logits + src * (long long)V;

    float bv = -__builtin_inff();
    int   bi = 0;

#define DDTS_CMP4(v, i0)                                                    \
    do {                                                                    \
        if ((v).x > bv) { bv = (v).x; bi = (i0);     }                      \
        if ((v).y > bv) { bv = (v).y; bi = (i0) + 1; }                      \
        if ((v).z > bv) { bv = (v).z; bi = (i0) + 2; }                      \
        if ((v).w > bv) { bv = (v).w; bi = (i0) + 3; }                      \
    } while (0)

    if ((V & 3) == 0) {
        const float4* b4  = (const float4*)base;
        const int     nf4 = V >> 2;

#if USE_ASYNC_LDS
        // 4-deep async ring buffer: each lane DMAs its own 16B slot, up to 3
        // chunks issued ahead of the one being consumed. s_wait_asynccnt is
        // per-wave and async loads complete in order, so waiting <=3 after
        // issuing chunk c+3 guarantees chunk c has landed in LDS. No
        // cross-wave barriers needed inside the streaming loop.
        __shared__ float4 stage[4][256];
        const int nc = nf4 / 256;   // full block-wide chunks of 256 float4s

        // prologue: issue chunks 0..min(nc,3)-1
        for (int p = 0; p < 3 && p < nc; ++p) {
            GASYNC_B128(b4 + p * 256 + tid, &stage[p][tid]);
        }

        int c = 0;
        // steady state: keep 3 chunks in flight
        for (; c + 3 < nc; ++c) {
            GASYNC_B128(b4 + (c + 3) * 256 + tid, &stage[(c + 3) & 3][tid]);
            asm volatile("s_wait_asynccnt 0x3" ::: "memory");
            const float4 v  = stage[c & 3][tid];
            const int    i0 = (c * 256 + tid) << 2;
            DDTS_CMP4(v, i0);
        }
        // drain: no more issues; pending = nc - 1 - c
        for (; c < nc; ++c) {
            switch (nc - 1 - c) {
                case 2:  asm volatile("s_wait_asynccnt 0x2" ::: "memory"); break;
                case 1:  asm volatile("s_wait_asynccnt 0x1" ::: "memory"); break;
                default: asm volatile("s_wait_asynccnt 0x0" ::: "memory"); break;
            }
            const float4 v  = stage[c & 3][tid];
            const int    i0 = (c * 256 + tid) << 2;
            DDTS_CMP4(v, i0);
        }
        // leftover float4s (nf4 % 256), strided directly
        for (int i = nc * 256 + tid; i < nf4; i += 256) {
            const float4 v = b4[i];
            DDTS_CMP4(v, i << 2);
        }
#else
        for (int i = tid; i < nf4; i += 256) {
            if (i + 2048 < nf4) __builtin_prefetch(b4 + i + 2048, 0, 0);
            const float4 v = b4[i];
            DDTS_CMP4(v, i << 2);
        }
#endif
    } else {
        for (int i = tid; i < V; i += 256) {
            const float v = base[i];
            if (v > bv) { bv = v; bi = i; }
        }
    }

    // scalar tail for V % 4 (only matters if V not divisible by 4)
    if (tid == 0) {
        for (int e = (V & ~3); e < V; ++e) {
            const float v = base[e];
            if (v > bv) { bv = v; bi = e; }
        }
    }

    // block reduction (max value, smallest index on ties)
    __shared__ float rv[256];
    __shared__ int   ri[256];
    rv[tid] = bv; ri[tid] = bi;
    __syncthreads();
    for (int s = 128; s > 0; s >>= 1) {
        if (tid < s) {
            const float ov = rv[tid + s];
            const int   oi = ri[tid + s];
            if (ov > rv[tid] || (ov == rv[tid] && oi < ri[tid])) {
                rv[tid] = ov; ri[tid] = oi;
            }
        }
        __syncthreads();
    }
    if (tid == 0) posterior[row] = ri[0];
}

// ---------------------------------------------------------------------------
// Kernel 2: per-request greedy tree traversal + output packing.
// Outputs (flat, in reference return order, as float per harness contract):
//   out_tok [B*N], acc_idx [B*(depth+1)], acc_len [B], next_ids [B], bonus [B]
// ---------------------------------------------------------------------------
__global__ void ddts_traverse(const int* __restrict__ posterior,
                              const int* __restrict__ tokens,
                              const int* __restrict__ children,
                              float*     __restrict__ out,
                              int B, int N, int K, int depth)
{
    const int b = threadIdx.x + blockIdx.x * blockDim.x;
    if (b >= B) return;

    float* out_tok  = out;
    float* acc_idx  = out_tok + (size_t)B * N;
    float* acc_len  = acc_idx + (size_t)B * (depth + 1);
    float* next_ids = acc_len + B;
    float* bonus    = next_ids + B;

    const int* post = posterior + (size_t)b * N;
    const int* tok  = tokens    + (size_t)b * N;

    int  tail[32];
    int  cur   = 0;
    int  count = 0;
    bool done  = false;

    for (int d = 0; d < depth; ++d) {
        int nxt = -1;
        if (!done) {
            const int t = post[cur];
            for (int j = 0; j < K; ++j) {
                const int ch = children[cur * K + j];
                if (ch >= 0 && tok[ch] == t) { nxt = ch; break; }  // first match
            }
        }
        tail[d] = nxt;
        if (nxt >= 0) { cur = nxt; ++count; }
        else          { done = true; }
    }

    const int accl  = 1 + count;
    const int fin   = cur;
    const int nextt = post[fin];

    float* orow = out_tok + (size_t)b * N;
    for (int i = 0; i < N; ++i) orow[i] = -1.0f;
    for (int d = 0; d < depth; ++d)
        if (tail[d] >= 0) orow[d] = (float)tok[tail[d]];
    orow[accl - 1] = (float)nextt;

    float* arow = acc_idx + (size_t)b * (depth + 1);
    arow[0] = 0.0f;
    for (int d = 0; d < depth; ++d) arow[d + 1] = (float)tail[d];

    acc_len[b]  = (float)accl;
    next_ids[b] = (float)nextt;
    bonus[b]    = (float)(b * N + fin);
}

// ---------------------------------------------------------------------------
extern "C" void kernel_launch(void* const* d_in, const int* in_sizes, int n_in,
                              void* d_out, int out_size, void* d_ws, size_t ws_size,
                              hipStream_t stream)
{
    const float* logits   = (const float*)d_in[0];
    const int*   tgt_idx  = (const int*)  d_in[1];
    const int*   tokens   = (const int*)  d_in[2];
    const int*   children = (const int*)  d_in[3];

    const int total = in_sizes[1];          // B * N rows
    const int Kc    = 2;                    // binary verification tree (reference)
    const int N     = in_sizes[3] / Kc;     // children is [N, K]
    const int B     = total / N;
    const int V     = (int)((long long)in_sizes[0] / (long long)total);

    int depth = 0;                          // log2(N+1) - 1
    for (int t = N + 1; t > 2; t >>= 1) ++depth;
    if (depth > 31) depth = 31;

    int* posterior = (int*)d_ws;            // total int32s of scratch

    ddts_argmax_rows<<<total, 256, 0, stream>>>(logits, tgt_idx, posterior, V);

    const int tb = 64;
    ddts_traverse<<<(B + tb - 1) / tb, tb, 0, stream>>>(
        posterior, tokens, children, (float*)d_out, B, N, Kc, depth);
}